// TimeSeriesAlignment_70068096467359
// MI455X (gfx1250) — compile-verified
//
#include <hip/hip_runtime.h>
#include <hip/hip_bf16.h>

typedef float v2f __attribute__((ext_vector_type(2)));
typedef float v8f __attribute__((ext_vector_type(8)));

#define BDIM 4
#define SDIM 512
#define HDIM 128

// ---------------------------------------------------------------------------
// GEMM: Y[M=2048, 128] = op(X[2048,128] @ W[128,128] + bias), via
// V_WMMA_F32_16X16X4_F32. One wave per 16x16 output tile, 8 waves / block.
// 1024 tiles -> grid of 128 blocks of 256 threads. Exact tiling, EXEC all-1s.
// ---------------------------------------------------------------------------
__global__ __launch_bounds__(256)
void gemm_h128_wmma_f32(const float* __restrict__ X,
                        const float* __restrict__ W,
                        const float* __restrict__ bias,
                        float* __restrict__ Y,
                        int relu_flag)
{
    const int lane = threadIdx.x & 31;
    const int wave = threadIdx.x >> 5;
    const int tile = blockIdx.x * 8 + wave;   // 0..1023
    const int tm   = tile >> 3;               // 0..127 (row tile of 16)
    const int tn   = tile & 7;                // 0..7   (col tile of 16)

    const int mrow = tm * 16 + (lane & 15);   // A-operand row for this lane
    const int ncol = tn * 16 + (lane & 15);   // B-operand col for this lane
    const int kb   = (lane >> 4) * 2;         // upper half-wave holds K+2,K+3

    const float* xrow = X + mrow * HDIM;

    v8f acc = {};
    #pragma unroll
    for (int k0 = 0; k0 < HDIM; k0 += 4) {
        v2f av, bv;
        av.x = xrow[k0 + kb + 0];
        av.y = xrow[k0 + kb + 1];
        bv.x = W[(k0 + kb + 0) * HDIM + ncol];
        bv.y = W[(k0 + kb + 1) * HDIM + ncol];
        acc = __builtin_amdgcn_wmma_f32_16x16x4_f32(
            /*neg_a=*/false, av, /*neg_b=*/false, bv,
            /*c_mod=*/(short)0, acc, /*reuse_a=*/false, /*reuse_b=*/false);
    }

    // C/D layout: VGPR r -> (M = tm*16 + r + 8*(lane>=16), N = tn*16 + lane&15)
    const int srow = tm * 16 + (lane >> 4) * 8;
    const float bcol = bias ? bias[ncol] : 0.0f;
    #pragma unroll
    for (int r = 0; r < 8; ++r) {
        float v = acc[r] + bcol;
        if (relu_flag) v = fmaxf(v, 0.0f);
        Y[(srow + r) * HDIM + ncol] = v;
    }
}

// ---------------------------------------------------------------------------
// Epilogue: one wave32 per (b, t). Computes the three needed diagonals of A
// (never forming the (B,S,S,H) tensor), P = log_softmax over 3 logits,
// argmax op, and the blended output row. Lane-strided over H=128 (4/lane),
// shfl-xor wave reduction.
// ---------------------------------------------------------------------------
__global__ __launch_bounds__(256)
void align_epilogue(const float* __restrict__ x,
                    const float* __restrict__ ref,
                    const float* __restrict__ a,
                    const float* __restrict__ c,
                    const float* __restrict__ h1,
                    const float* __restrict__ b1,
                    const float* __restrict__ W2,
                    const float* __restrict__ b2,
                    const float* __restrict__ Wp2,
                    const float* __restrict__ bp2,
                    float* __restrict__ out)
{
    const int lane  = threadIdx.x & 31;
    const int gwave = (blockIdx.x * blockDim.x + threadIdx.x) >> 5; // 0..2047
    const int b = gwave / SDIM;
    const int t = gwave % SDIM;

    const int rowt = (b * SDIM + t) * HDIM;

    if (t == 0) {
        // out[:, 0] = x[:, 0]
        #pragma unroll
        for (int i = 0; i < 4; ++i) {
            int h = lane + 32 * i;
            out[rowt + h] = x[rowt + h];
        }
        return;
    }

    const int rowt1 = rowt - HDIM; // (b, t-1)

    const float* at  = a + rowt;
    const float* at1 = a + rowt1;
    const float* ct  = c + rowt;
    const float* ct1 = c + rowt1;
    const float* h1t = h1 + rowt;

    float dDiag = 0.f, dSup = 0.f, dSub = 0.f;   // A[t,t], A[t-1,t], A[t,t-1]
    float l0 = 0.f, l1 = 0.f, l2 = 0.f;          // logits for P

    #pragma unroll
    for (int i = 0; i < 4; ++i) {
        int h = lane + 32 * i;
        float w2 = W2[h];
        float bb = b1[h];
        float av = at[h], a1 = at1[h], cv = ct[h], c1 = ct1[h];
        dDiag += fmaxf(av + cv + bb, 0.f) * w2;
        dSup  += fmaxf(a1 + cv + bb, 0.f) * w2;
        dSub  += fmaxf(av + c1 + bb, 0.f) * w2;
        float hv = h1t[h];
        l0 += hv * Wp2[h * 3 + 0];
        l1 += hv * Wp2[h * 3 + 1];
        l2 += hv * Wp2[h * 3 + 2];
    }

    // wave32 butterfly reduction -> every lane holds the totals
    #pragma unroll
    for (int off = 16; off > 0; off >>= 1) {
        dDiag += __shfl_xor(dDiag, off, 32);
        dSup  += __shfl_xor(dSup,  off, 32);
        dSub  += __shfl_xor(dSub,  off, 32);
        l0    += __shfl_xor(l0,    off, 32);
        l1    += __shfl_xor(l1,    off, 32);
        l2    += __shfl_xor(l2,    off, 32);
    }

    const float b2v = b2[0];
    float Ad = 1.f / (1.f + expf(-(dDiag + b2v)));  // diag -> alpha
    float As = 1.f / (1.f + expf(-(dSup  + b2v)));
    float Ab = 1.f / (1.f + expf(-(dSub  + b2v)));

    l0 += bp2[0]; l1 += bp2[1]; l2 += bp2[2];
    float mx  = fmaxf(l0, fmaxf(l1, l2));
    float lse = mx + logf(expf(l0 - mx) + expf(l1 - mx) + expf(l2 - mx));
    float P0 = l0 - lse, P1 = l1 - lse, P2 = l2 - lse;

    float s0 = Ad * P0, s1 = As * P1, s2 = Ab * P2;
    int op = 0; float best = s0;                // first-max tie-break like argmax
    if (s1 > best) { best = s1; op = 1; }
    if (s2 > best) { best = s2; op = 2; }

    #pragma unroll
    for (int i = 0; i < 4; ++i) {
        int h = lane + 32 * i;
        float v;
        if (op == 0)      v = (1.f - Ad) * x[rowt + h] + Ad * ref[rowt + h];
        else if (op == 1) v = ref[rowt + h];
        else              v = x[rowt1 + h];
        out[rowt + h] = v;
    }
}

extern "C" void kernel_launch(void* const* d_in, const int* in_sizes, int n_in,
                              void* d_out, int out_size, void* d_ws, size_t ws_size,
                              hipStream_t stream) {
    const float* x   = (const float*)d_in[0];
    const float* ref = (const float*)d_in[1];
    const float* Wa  = (const float*)d_in[2];
    const float* Wb  = (const float*)d_in[3];
    const float* b1  = (const float*)d_in[4];
    const float* W2  = (const float*)d_in[5];
    const float* b2  = (const float*)d_in[6];
    const float* Wp1 = (const float*)d_in[7];
    const float* bp1 = (const float*)d_in[8];
    const float* Wp2 = (const float*)d_in[9];
    const float* bp2 = (const float*)d_in[10];
    float* out = (float*)d_out;

    const int NROW = BDIM * SDIM * HDIM; // 262144 floats per (B,S,H) tensor
    float* a_ws  = (float*)d_ws;
    float* c_ws  = a_ws + NROW;
    float* h1_ws = a_ws + 2 * NROW;

    dim3 blk(256);
    // (2048/16)*(128/16) = 1024 tiles, 8 waves/block -> 128 blocks
    dim3 gemm_grid(128);
    gemm_h128_wmma_f32<<<gemm_grid, blk, 0, stream>>>(x,   Wa,  nullptr, a_ws,  0);
    gemm_h128_wmma_f32<<<gemm_grid, blk, 0, stream>>>(ref, Wb,  nullptr, c_ws,  0);
    gemm_h128_wmma_f32<<<gemm_grid, blk, 0, stream>>>(x,   Wp1, bp1,     h1_ws, 1);

    // B*S = 2048 waves, 8 waves/block -> 256 blocks
    dim3 epi_grid(256);
    align_epilogue<<<epi_grid, blk, 0, stream>>>(x, ref, a_ws, c_ws, h1_ws,
                                                 b1, W2, b2, Wp2, bp2, out);
}